// GCNsimple_12077448036413
// MI455X (gfx1250) — compile-verified
//
#include <hip/hip_runtime.h>
#include <hip/hip_bf16.h>

// ---------------------------------------------------------------------------
// GCN (2-layer) for MI455X / gfx1250, wave32.
//   h1  = x @ W1                       (f32 WMMA 16x16x4)
//   a1  = b1 + scatter(norm * h1[src]) (self-loop folded into init, atomics)
//   h2  = relu(a1) @ W2                (ReLU fused into A-load of WMMA GEMM)
//   out = b2 + scatter(norm * h2[src])
// ---------------------------------------------------------------------------

typedef __attribute__((ext_vector_type(2))) float v2f;
typedef __attribute__((ext_vector_type(8))) float v8f;

// ---------------- degree / normalization ----------------

__global__ void zero_f32_kernel(float* __restrict__ p, int n) {
  int i = blockIdx.x * blockDim.x + threadIdx.x;
  if (i < n) p[i] = 0.0f;
}

__global__ void deg_count_kernel(const int* __restrict__ dst,
                                 float* __restrict__ deg, int E) {
  int e = blockIdx.x * blockDim.x + threadIdx.x;
  if (e < E) atomicAdd(&deg[dst[e]], 1.0f);
}

// deg -> dinv = 1/sqrt(deg + 1)   (+1 = self loop, so always > 0)
__global__ void dinv_kernel(float* __restrict__ deg, int n) {
  int i = blockIdx.x * blockDim.x + threadIdx.x;
  if (i < n) deg[i] = 1.0f / sqrtf(deg[i] + 1.0f);
}

// ---------------- f32 WMMA GEMM:  C[M,N] = op(A[M,K]) @ B[K,N] ----------------
// One 16x16 output tile per wave; block carries N/16 waves (all N tiles of one
// M tile). K stepped by 4 per V_WMMA_F32_16X16X4_F32. M must be multiple of 16
// (100000 = 6250*16) so EXEC is all-ones at every WMMA as the ISA requires.
//
// A 16x4 layout : lanes 0-15 -> M=lane, K={k0,k0+1}; lanes 16-31 -> K={k0+2,k0+3}
// B 4x16 layout : lanes 0-15 -> N=lane, K rows {k0,k0+1}; lanes 16-31 -> {k0+2,k0+3}
// C/D 16x16     : VGPR r, lanes 0-15 -> M=r, lanes 16-31 -> M=r+8; N=lane&15
template <bool RELU_A>
__global__ void wmma_gemm_f32_kernel(const float* __restrict__ A,
                                     const float* __restrict__ B,
                                     float* __restrict__ C,
                                     int M, int N, int K) {
  const int wave = threadIdx.x >> 5;   // N-tile index within block
  const int lane = threadIdx.x & 31;
  const int hi   = lane >> 4;          // 0: lanes 0-15, 1: lanes 16-31
  const int l    = lane & 15;

  const int m0 = blockIdx.x * 16;
  const int n0 = wave * 16;

  const long arow = (long)(m0 + l) * K;     // A row base for this lane
  const int  bcol = n0 + l;                 // B column for this lane

  v8f acc = (v8f)0.0f;

  for (int k0 = 0; k0 < K; k0 += 4) {
    const int ka = k0 + hi * 2;
    v2f a;
    a.x = A[arow + ka];
    a.y = A[arow + ka + 1];
    if (RELU_A) {
      a.x = fmaxf(a.x, 0.0f);
      a.y = fmaxf(a.y, 0.0f);
    }
    v2f b;
    b.x = B[(long)ka * N + bcol];
    b.y = B[(long)(ka + 1) * N + bcol];
    acc = __builtin_amdgcn_wmma_f32_16x16x4_f32(
        /*neg_a=*/false, a, /*neg_b=*/false, b,
        /*c_mod=*/(short)0, acc, /*reuse_a=*/false, /*reuse_b=*/false);
  }

#pragma unroll
  for (int r = 0; r < 8; ++r) {
    const int row = m0 + r + hi * 8;
    C[(long)row * N + n0 + l] = acc[r];
  }
}

// ---------------- aggregation ----------------

// agg[i,f] = bias[f] + dinv[i]^2 * h[i,f]   (self-loop term + bias, also zeroes)
__global__ void agg_init_kernel(const float* __restrict__ h,
                                const float* __restrict__ dinv,
                                const float* __restrict__ bias,
                                float* __restrict__ agg, int F) {
  const int i = blockIdx.x;
  const int f = threadIdx.x;
  const float d = dinv[i];
  agg[(long)i * F + f] = fmaf(d * d, h[(long)i * F + f], bias[f]);
}

// agg[dst,f] += dinv[src]*dinv[dst] * h[src,f]; F/4 lanes per edge, float4 gather.
template <int F>
__global__ void edge_scatter_kernel(const int* __restrict__ src,
                                    const int* __restrict__ dst,
                                    const float* __restrict__ dinv,
                                    const float* __restrict__ h,
                                    float* __restrict__ agg, int E) {
  constexpr int LPE = F / 4;  // lanes per edge
  const int gid = blockIdx.x * blockDim.x + threadIdx.x;
  const int e   = gid / LPE;
  const int f4  = (gid % LPE) * 4;
  if (e >= E) return;
  const int s = src[e];
  const int d = dst[e];
  const float norm = dinv[s] * dinv[d];
  const float4 hv = *reinterpret_cast<const float4*>(h + (long)s * F + f4);
  float* ap = agg + (long)d * F + f4;
  atomicAdd(ap + 0, norm * hv.x);
  atomicAdd(ap + 1, norm * hv.y);
  atomicAdd(ap + 2, norm * hv.z);
  atomicAdd(ap + 3, norm * hv.w);
}

// ---------------- driver ----------------

extern "C" void kernel_launch(void* const* d_in, const int* in_sizes, int n_in,
                              void* d_out, int out_size, void* d_ws, size_t ws_size,
                              hipStream_t stream) {
  constexpr int D_IN = 128, D_HID = 128, D_OUT = 64;

  const float* x  = (const float*)d_in[0];
  const int* eidx = (const int*)d_in[1];
  const float* W1 = (const float*)d_in[2];
  const float* b1 = (const float*)d_in[3];
  const float* W2 = (const float*)d_in[4];
  const float* b2 = (const float*)d_in[5];
  float* out = (float*)d_out;

  const int Nn = in_sizes[0] / D_IN;   // 100000 nodes
  const int E  = in_sizes[1] / 2;      // 1600000 edges
  const int* src = eidx;
  const int* dst = eidx + E;

  // workspace layout
  char* ws = (char*)d_ws;
  float* dinv = (float*)ws;                                  // Nn floats
  size_t off = ((size_t)Nn * sizeof(float) + 255) & ~(size_t)255;
  float* hbuf = (float*)(ws + off);                          // Nn * 128 floats
  off += ((size_t)Nn * D_HID * sizeof(float) + 255) & ~(size_t)255;
  float* aggbuf = (float*)(ws + off);                        // Nn * 128 floats

  // 1. degrees -> dinv
  zero_f32_kernel<<<(Nn + 255) / 256, 256, 0, stream>>>(dinv, Nn);
  deg_count_kernel<<<(E + 255) / 256, 256, 0, stream>>>(dst, dinv, E);
  dinv_kernel<<<(Nn + 255) / 256, 256, 0, stream>>>(dinv, Nn);

  // 2. h1 = x @ W1  (8 N-tiles -> 8 waves = 256 threads per block)
  wmma_gemm_f32_kernel<false><<<Nn / 16, (D_HID / 16) * 32, 0, stream>>>(
      x, W1, hbuf, Nn, D_HID, D_IN);

  // 3. a1 = b1 + dinv^2*h1, then edge scatter (1 wave per edge: 32 lanes * 4f)
  agg_init_kernel<<<Nn, D_HID, 0, stream>>>(hbuf, dinv, b1, aggbuf, D_HID);
  {
    const long threads = (long)E * (D_HID / 4);
    edge_scatter_kernel<D_HID><<<(unsigned)((threads + 255) / 256), 256, 0, stream>>>(
        src, dst, dinv, hbuf, aggbuf, E);
  }

  // 4. h2 = relu(a1) @ W2  (ReLU fused into A load; 4 waves = 128 threads)
  wmma_gemm_f32_kernel<true><<<Nn / 16, (D_OUT / 16) * 32, 0, stream>>>(
      aggbuf, W2, hbuf, Nn, D_OUT, D_HID);

  // 5. out = b2 + dinv^2*h2, then edge scatter (16 lanes per edge)
  agg_init_kernel<<<Nn, D_OUT, 0, stream>>>(hbuf, dinv, b2, out, D_OUT);
  {
    const long threads = (long)E * (D_OUT / 4);
    edge_scatter_kernel<D_OUT><<<(unsigned)((threads + 255) / 256), 256, 0, stream>>>(
        src, dst, dinv, hbuf, out, E);
  }
}